// Anomalous_regression_63806034149939
// MI455X (gfx1250) — compile-verified
//
#include <hip/hip_runtime.h>

// ConvLSTM1D fused kernel for MI455X (gfx1250, wave32, WMMA).
// Per time step: one fused GEMM  z[m,n] = [x_row(16) | h_row(64)] @ W(80x128) + bias
// computed with V_WMMA_F32_16X16X4_F32 (fp32 matrix core), then LSTM epilogue.

typedef float v2f __attribute__((ext_vector_type(2)));
typedef float v8f __attribute__((ext_vector_type(8)));

#define B_  32
#define T_  50
#define S_  2048
#define C_  8
#define F_  32
#define SH  1024          // S / stride
#define NG  128           // 4*F gate channels
#define KTOT 80           // 16 (input conv) + 64 (recurrent conv)
#define HSTRIDE (SH + 1)  // one zero pad row per batch for h[p+1] at p=SH-1

__device__ __forceinline__ float hsig(float x) {
    return fminf(fmaxf(0.2f * x + 0.5f, 0.0f), 1.0f);
}

__global__ void init_state(float* __restrict__ h0, float* __restrict__ h1,
                           float* __restrict__ c) {
    size_t n = (size_t)B_ * HSTRIDE * F_;
    size_t i = (size_t)blockIdx.x * blockDim.x + threadIdx.x;
    if (i < n) { h0[i] = 0.0f; h1[i] = 0.0f; c[i] = 0.0f; }
}

// Block: 256 threads (8 waves). Each block handles 16 consecutive rows m = b*SH + p
// (SH divisible by 16 => all rows same batch). Wave w computes the 16x16 tile of
// columns [16w, 16w+16).
__global__ __launch_bounds__(256) void lstm_step(
    const float* __restrict__ x,    // (B, T, S, C)
    const float* __restrict__ wk,   // (2, 8, 128)  == rows 0..15 of W
    const float* __restrict__ wr,   // (2, 32, 128) == rows 16..79 of W
    const float* __restrict__ bias, // (128)
    const float* __restrict__ h_in, // (B, HSTRIDE, F), pad row zeroed
    float* __restrict__ h_out,      // (B, HSTRIDE, F)
    float* __restrict__ c_st,       // (B, HSTRIDE, F), updated in place
    int t)
{
    __shared__ float sW[KTOT * NG];   // 80x128 stacked weights (40 KB)
    __shared__ float sA[16 * 81];     // 16 rows x 80 (padded to 81 vs bank conflicts)
    __shared__ float sZ[16 * NG];     // gate pre-activations z

    const int tid = threadIdx.x;
    const int m0  = blockIdx.x * 16;

    // Stage stacked weight matrix W (80x128) into LDS.
    for (int i = tid; i < 16 * NG; i += 256) sW[i] = wk[i];
    for (int i = tid; i < 64 * NG; i += 256) sW[16 * NG + i] = wr[i];

    // Stage A rows. x part: row r gets x[b, t, 2p : 2p+2, :] = 16 contiguous floats.
    {
        int r = tid >> 4, j = tid & 15;
        int m = m0 + r; int b = m >> 10; int p = m & (SH - 1);
        sA[r * 81 + j] = x[(((size_t)b * T_ + t) * S_ + 2 * p) * C_ + j];
    }
    // h part: row r gets h[b, p : p+2, :] = 64 contiguous floats (pad row is zero).
    for (int q = 0; q < 4; ++q) {
        int idx = tid + q * 256;          // 0..1023
        int r = idx >> 6, j = idx & 63;
        int m = m0 + r; int b = m >> 10; int p = m & (SH - 1);
        sA[r * 81 + 16 + j] = h_in[((size_t)b * HSTRIDE + p) * F_ + j];
    }
    __syncthreads();

    const int wave = tid >> 5;          // n-tile index 0..7
    const int lane = tid & 31;
    const int half = lane >> 4;         // 0: lanes 0-15, 1: lanes 16-31
    const int l16  = lane & 15;
    const int n    = wave * 16 + l16;   // global gate column

    // K=80 via 20 chained fp32 WMMAs (16x16x4).
    // A frag (ISA layout): lanes 0-15 hold {K=k0,k0+1}, lanes 16-31 hold {K=k0+2,k0+3}.
    v8f acc = {};
    for (int i = 0; i < 20; ++i) {
        int k = 4 * i + 2 * half;
        v2f a, b;
        a.x = sA[l16 * 81 + k];
        a.y = sA[l16 * 81 + k + 1];
        b.x = sW[k * NG + n];
        b.y = sW[(k + 1) * NG + n];
        acc = __builtin_amdgcn_wmma_f32_16x16x4_f32(
            /*neg_a=*/false, a, /*neg_b=*/false, b,
            /*c_mod=*/(short)0, acc, /*reuse_a=*/false, /*reuse_b=*/false);
    }

    // Scatter D (16x16 f32 layout: VGPR v -> M = v + 8*half, N = l16) into LDS + bias.
    float bn = bias[n];
    for (int v = 0; v < 8; ++v) {
        int row = half ? (v + 8) : v;
        sZ[row * NG + n] = acc[v] + bn;
    }
    __syncthreads();

    // LSTM epilogue: 16 rows x 32 features, gates live at n = f, 32+f, 64+f, 96+f.
    for (int e = tid; e < 16 * F_; e += 256) {
        int r = e >> 5, f = e & 31;
        int m = m0 + r; int b = m >> 10; int p = m & (SH - 1);
        size_t sidx = ((size_t)b * HSTRIDE + p) * F_ + f;
        float zi = sZ[r * NG + f];
        float zf = sZ[r * NG + 32 + f];
        float zc = sZ[r * NG + 64 + f];
        float zo = sZ[r * NG + 96 + f];
        float ig = hsig(zi), fg = hsig(zf), og = hsig(zo);
        float cn = fg * c_st[sidx] + ig * tanhf(zc);
        c_st[sidx]  = cn;
        h_out[sidx] = og * tanhf(cn);
    }
}

// out[b] = sum_{p,f} h[b,p,f] * w[p*F + f] + bconst
__global__ __launch_bounds__(256) void dense_out(
    const float* __restrict__ h, const float* __restrict__ w,
    const float* __restrict__ bconst, float* __restrict__ out)
{
    __shared__ float red[256];
    int b = blockIdx.x;
    float s = 0.0f;
    for (int i = threadIdx.x; i < SH * F_; i += 256) {
        int p = i >> 5, f = i & 31;
        s += h[((size_t)b * HSTRIDE + p) * F_ + f] * w[i];
    }
    red[threadIdx.x] = s;
    __syncthreads();
    for (int o = 128; o > 0; o >>= 1) {
        if (threadIdx.x < o) red[threadIdx.x] += red[threadIdx.x + o];
        __syncthreads();
    }
    if (threadIdx.x == 0) out[b] = red[0] + bconst[0];
}

extern "C" void kernel_launch(void* const* d_in, const int* in_sizes, int n_in,
                              void* d_out, int out_size, void* d_ws, size_t ws_size,
                              hipStream_t stream) {
    const float* x    = (const float*)d_in[0];
    const float* wk   = (const float*)d_in[1];
    const float* wr   = (const float*)d_in[2];
    const float* bias = (const float*)d_in[3];
    const float* dw   = (const float*)d_in[4];
    const float* db   = (const float*)d_in[5];
    float* out = (float*)d_out;

    const size_t stateN = (size_t)B_ * HSTRIDE * F_;
    float* h0 = (float*)d_ws;
    float* h1 = h0 + stateN;
    float* c  = h1 + stateN;

    init_state<<<(int)((stateN + 255) / 256), 256, 0, stream>>>(h0, h1, c);

    float* hin = h0; float* hout = h1;
    for (int t = 0; t < T_; ++t) {
        lstm_step<<<(B_ * SH) / 16, 256, 0, stream>>>(x, wk, wr, bias, hin, hout, c, t);
        float* tmp = hin; hin = hout; hout = tmp;
    }
    dense_out<<<B_, 256, 0, stream>>>(hin, dw, db, out);
}